// SPDNetWithNorm_22643067584708
// MI455X (gfx1250) — compile-verified
//
#include <hip/hip_runtime.h>

#define BATCH 2048
#define NCH 128
#define D 64
#define LDA 65           // padded LDS row stride (64 banks -> conflict-free rows & cols)
#define NSWEEP 8
#define TRACE_EPS 1e-4f
#define REEIG_THR 1e-4f
#define BN_EPS 1e-5f
#define SQRT2F 1.4142135623730951f

typedef float v2f __attribute__((ext_vector_type(2)));
typedef float v8f __attribute__((ext_vector_type(8)));

// ---------------------------------------------------------------------------
// WMMA helpers: fp32 16x16 tiles via V_WMMA_F32_16X16X4_F32
//   A frag (16x4): lanes 0-15 -> K={k0,k0+1}, lanes 16-31 -> K={k0+2,k0+3}
//   B frag (4x16): mirrored, N = lane&15
//   C/D (16x16):   vgpr r, row = r + 8*(lane>=16), col = lane&15
// ---------------------------------------------------------------------------
__device__ __forceinline__ void store_tile(float* Dm, int ld, int mB, int nB, v8f acc) {
  const int lane = threadIdx.x & 31;
  const int col  = nB + (lane & 15);
  const int row0 = mB + ((lane >> 4) << 3);
#pragma unroll
  for (int r = 0; r < 8; ++r) Dm[(row0 + r) * ld + col] = acc[r];
}

template<bool AT, bool BT, bool SC>
__device__ __forceinline__ v8f wmma_tileK(const float* A, int lda, const float* B, int ldb,
                                          int mB, int nB, int K, const float* ks, v8f acc) {
  const int lane = threadIdx.x & 31;
  const int mr   = lane & 15;
  const int kh   = (lane >> 4) << 1;
  for (int k0 = 0; k0 < K; k0 += 4) {
    const int k = k0 + kh;
    v2f a, b;
    if (AT) { a.x = A[k * lda + mB + mr];       a.y = A[(k + 1) * lda + mB + mr]; }
    else    { a.x = A[(mB + mr) * lda + k];     a.y = A[(mB + mr) * lda + k + 1]; }
    if (BT) { b.x = B[(nB + mr) * ldb + k];     b.y = B[(nB + mr) * ldb + k + 1]; }
    else    { b.x = B[k * ldb + nB + mr];       b.y = B[(k + 1) * ldb + nB + mr]; }
    if (SC) { b.x *= ks[k];                     b.y *= ks[k + 1]; }
    acc = __builtin_amdgcn_wmma_f32_16x16x4_f32(false, a, false, b, (short)0, acc,
                                                false, false);
  }
  return acc;
}

// Full 64x64 = 64x64 * 64x64 with 8 waves (2 tiles each), LDS stride LDA.
// If BT/SC: B is used as diag(ks) * B^T (for V * diag(f) * V^T reconstructions).
template<bool AT, bool BT, bool SC>
__device__ __forceinline__ void mm64(const float* A, const float* B, const float* ks,
                                     float* Dlds, float* Dglb) {
  const int wave = threadIdx.x >> 5;
#pragma unroll
  for (int i = 0; i < 2; ++i) {
    const int t  = wave * 2 + i;
    const int mB = (t >> 2) << 4, nB = (t & 3) << 4;
    v8f acc = {0.f, 0.f, 0.f, 0.f, 0.f, 0.f, 0.f, 0.f};
    acc = wmma_tileK<AT, BT, SC>(A, LDA, B, LDA, mB, nB, D, ks, acc);
    if (Dlds) store_tile(Dlds, LDA, mB, nB, acc);
    else      store_tile(Dglb, D, mB, nB, acc);
  }
}

// ---------------------------------------------------------------------------
// Parallel cyclic Jacobi eigensolver for symmetric 64x64 in LDS (256 threads).
// Tournament ordering: 63 rounds of 32 disjoint rotations per sweep.
// On exit: diag(A) = eigenvalues, V columns = eigenvectors (A0 = V diag Vt).
// ---------------------------------------------------------------------------
struct JScratch { float c[32]; float s[32]; int p[32]; int q[32]; int perm[64]; };

__device__ void jacobi64(float* A, float* V, JScratch* js) {
  const int tid = threadIdx.x;
  for (int e = tid; e < D * D; e += 256) {
    int i = e >> 6, j = e & 63;
    V[i * LDA + j] = (i == j) ? 1.f : 0.f;
  }
  if (tid < 64) js->perm[tid] = tid;
  __syncthreads();
  for (int sweep = 0; sweep < NSWEEP; ++sweep) {
    for (int rnd = 0; rnd < 63; ++rnd) {
      if (tid < 32) {
        int p = js->perm[tid], q = js->perm[63 - tid];
        if (p > q) { int t = p; p = q; q = t; }
        float app = A[p * LDA + p], aqq = A[q * LDA + q], apq = A[p * LDA + q];
        float c = 1.f, s = 0.f;
        if (fabsf(apq) > 1e-12f) {
          float th = 0.5f * (aqq - app) / apq;
          float t  = copysignf(1.f, th) / (fabsf(th) + sqrtf(1.f + th * th));
          c = rsqrtf(1.f + t * t);
          s = t * c;
        }
        js->p[tid] = p; js->q[tid] = q; js->c[tid] = c; js->s[tid] = s;
      }
      __syncthreads();
      { // row phase: thread = (col j, pair-group g); rows disjoint across pairs
        const int j = tid & 63, g = tid >> 6;
        for (int k = g * 8; k < g * 8 + 8; ++k) {
          int p = js->p[k], q = js->q[k];
          float c = js->c[k], s = js->s[k];
          float ap = A[p * LDA + j], aq = A[q * LDA + j];
          A[p * LDA + j] = c * ap - s * aq;
          A[q * LDA + j] = s * ap + c * aq;
        }
      }
      __syncthreads();
      { // column phase on A and V
        const int j = tid & 63, g = tid >> 6;
        for (int k = g * 8; k < g * 8 + 8; ++k) {
          int p = js->p[k], q = js->q[k];
          float c = js->c[k], s = js->s[k];
          float ap = A[j * LDA + p], aq = A[j * LDA + q];
          A[j * LDA + p] = c * ap - s * aq;
          A[j * LDA + q] = s * ap + c * aq;
          float vp = V[j * LDA + p], vq = V[j * LDA + q];
          V[j * LDA + p] = c * vp - s * vq;
          V[j * LDA + q] = s * vp + c * vq;
        }
      }
      __syncthreads();
      int nv = 0;
      if (tid < 64) nv = js->perm[(tid == 0) ? 0 : ((tid == 1) ? 63 : tid - 1)];
      __syncthreads();
      if (tid < 64) js->perm[tid] = nv;
      __syncthreads();
    }
  }
}

__device__ __forceinline__ void load64(float* Alds, const float* g) {
  for (int e = threadIdx.x; e < D * D; e += 256)
    Alds[(e >> 6) * LDA + (e & 63)] = g[e];
}
__device__ __forceinline__ void store_sym64(float* g, const float* Alds) {
  for (int e = threadIdx.x; e < D * D; e += 256) {
    int i = e >> 6, j = e & 63;
    g[e] = 0.5f * (Alds[i * LDA + j] + Alds[j * LDA + i]);
  }
}

// ---------------------------------------------------------------------------
// Kernels
// ---------------------------------------------------------------------------

// WtW = W^T W (once), Bs = sym(bn_bias)^(1/2) via Jacobi + WMMA reconstruction.
__global__ __launch_bounds__(256) void k_prep(const float* __restrict__ W,
                                              const float* __restrict__ bnb,
                                              float* __restrict__ gWtW,
                                              float* __restrict__ gBs) {
  __shared__ float A[D * LDA]; __shared__ float V[D * LDA];
  __shared__ float f[D]; __shared__ JScratch js;
  const int tid = threadIdx.x;
  for (int e = tid; e < D * D; e += 256) {
    int i = e >> 6, j = e & 63;
    float s = 0.f;
    for (int k = 0; k < NCH; ++k) s += W[k * D + i] * W[k * D + j];
    gWtW[e] = s;
    A[i * LDA + j] = 0.5f * (bnb[i * D + j] + bnb[j * D + i]);
  }
  __syncthreads();
  jacobi64(A, V, &js);
  if (tid < 64) f[tid] = sqrtf(fmaxf(A[tid * LDA + tid], 0.f));
  __syncthreads();
  mm64<false, true, true>(V, V, f, nullptr, gBs);
}

// M_b = c1 * W^T X_b W + c2 * WtW  (trace-norm + shrinkage folded into scalars)
__global__ __launch_bounds__(256) void k_bimap(const float* __restrict__ Xg,
                                               const float* __restrict__ alphap,
                                               const float* __restrict__ Wg,
                                               const float* __restrict__ WtW,
                                               float* __restrict__ Mout) {
  __shared__ float Ws[NCH * LDA];   // W (128x64)
  __shared__ float Xs[NCH * 17];    // X column-block (128x16)
  __shared__ float Ts[D * 17];      // T = W^T * Xcb   (64x16)
  __shared__ float Ps[D * LDA];     // P = W^T X W     (64x64)
  __shared__ float red[256];
  __shared__ float cc[2];
  const int tid = threadIdx.x;
  const int b   = blockIdx.x;
  const float* Xb = Xg + (size_t)b * NCH * NCH;

  for (int e = tid; e < NCH * D; e += 256) Ws[(e >> 6) * LDA + (e & 63)] = Wg[e];
  red[tid] = (tid < NCH) ? Xb[tid * NCH + tid] : 0.f;
  __syncthreads();
  for (int s = 128; s > 0; s >>= 1) { if (tid < s) red[tid] += red[tid + s]; __syncthreads(); }
  if (tid == 0) {
    float tr = red[0], a = alphap[0];
    cc[0] = (1.f - a) / (tr + TRACE_EPS);
    cc[1] = a * tr / ((tr + TRACE_EPS) * (float)NCH);
  }
  __syncthreads();

  const int wave = tid >> 5;
  const int t0 = wave * 2, t1 = t0 + 1;
  const int m0 = (t0 >> 2) << 4, n0 = (t0 & 3) << 4;
  const int m1 = (t1 >> 2) << 4, n1 = (t1 & 3) << 4;
  v8f acc0 = {0.f,0.f,0.f,0.f,0.f,0.f,0.f,0.f};
  v8f acc1 = acc0;

  for (int cb = 0; cb < 8; ++cb) {
    for (int e = tid; e < NCH * 16; e += 256)
      Xs[(e >> 4) * 17 + (e & 15)] = Xb[(e >> 4) * NCH + cb * 16 + (e & 15)];
    __syncthreads();
    if (wave < 4) {   // T(64x16) = W^T(64x128) * Xcb(128x16), one tile per wave
      v8f t = {0.f,0.f,0.f,0.f,0.f,0.f,0.f,0.f};
      t = wmma_tileK<true, false, false>(Ws, LDA, Xs, 17, wave * 16, 0, NCH, nullptr, t);
      store_tile(Ts, 17, wave * 16, 0, t);
    }
    __syncthreads();
    // P += T(64x16) * W[cb*16 : cb*16+16, :](16x64)
    acc0 = wmma_tileK<false, false, false>(Ts, 17, Ws + cb * 16 * LDA, LDA, m0, n0, 16, nullptr, acc0);
    acc1 = wmma_tileK<false, false, false>(Ts, 17, Ws + cb * 16 * LDA, LDA, m1, n1, 16, nullptr, acc1);
    __syncthreads();
  }
  store_tile(Ps, LDA, m0, n0, acc0);
  store_tile(Ps, LDA, m1, n1, acc1);
  __syncthreads();
  const float c1 = cc[0], c2 = cc[1];
  float* Mb = Mout + (size_t)b * D * D;
  for (int e = tid; e < D * D; e += 256) {
    int i = e >> 6, j = e & 63;
    Mb[e] = c1 * 0.5f * (Ps[i * LDA + j] + Ps[j * LDA + i]) + c2 * WtW[e];
  }
}

// ReEig + logm sharing one eigendecomposition.
__global__ __launch_bounds__(256) void k_eig_reeig(const float* __restrict__ Min,
                                                   float* __restrict__ Xout,
                                                   float* __restrict__ Lout) {
  __shared__ float A[D * LDA]; __shared__ float V[D * LDA];
  __shared__ float f[D]; __shared__ JScratch js;
  const int tid = threadIdx.x; const int b = blockIdx.x;
  load64(A, Min + (size_t)b * D * D);
  __syncthreads();
  jacobi64(A, V, &js);
  if (tid < 64) f[tid] = fmaxf(A[tid * LDA + tid], REEIG_THR);
  __syncthreads();
  mm64<false, true, true>(V, V, f, nullptr, Xout + (size_t)b * D * D);
  __syncthreads();
  if (tid < 64) f[tid] = logf(f[tid]);
  __syncthreads();
  mm64<false, true, true>(V, V, f, nullptr, Lout + (size_t)b * D * D);
}

// logm of per-element SPD matrices.
__global__ __launch_bounds__(256) void k_eig_log(const float* __restrict__ Sin,
                                                 float* __restrict__ Lout) {
  __shared__ float A[D * LDA]; __shared__ float V[D * LDA];
  __shared__ float f[D]; __shared__ JScratch js;
  const int tid = threadIdx.x; const int b = blockIdx.x;
  load64(A, Sin + (size_t)b * D * D);
  __syncthreads();
  jacobi64(A, V, &js);
  if (tid < 64) f[tid] = logf(fmaxf(A[tid * LDA + tid], 1e-20f));
  __syncthreads();
  mm64<false, true, true>(V, V, f, nullptr, Lout + (size_t)b * D * D);
}

// Deterministic batch mean: one block per matrix element.
__global__ __launch_bounds__(256) void k_reduce_mean(const float* __restrict__ src,
                                                     float* __restrict__ dst) {
  __shared__ float red[256];
  const int tid = threadIdx.x; const int e = blockIdx.x;
  float s = 0.f;
  for (int b = tid; b < BATCH; b += 256) s += src[(size_t)b * D * D + e];
  red[tid] = s; __syncthreads();
  for (int k = 128; k > 0; k >>= 1) { if (tid < k) red[tid] += red[tid + k]; __syncthreads(); }
  if (tid == 0) dst[e] = red[0] * (1.f / (float)BATCH);
}

// G = expm(mean)
__global__ __launch_bounds__(256) void k_expm_small(const float* __restrict__ Min,
                                                    float* __restrict__ Gout) {
  __shared__ float A[D * LDA]; __shared__ float V[D * LDA];
  __shared__ float f[D]; __shared__ JScratch js;
  const int tid = threadIdx.x;
  load64(A, Min);
  __syncthreads();
  jacobi64(A, V, &js);
  if (tid < 64) f[tid] = expf(A[tid * LDA + tid]);
  __syncthreads();
  mm64<false, true, true>(V, V, f, nullptr, Gout);
}

// Gs = G^{1/2}, Gi = G^{-1/2}
__global__ __launch_bounds__(256) void k_G_pow(const float* __restrict__ G,
                                               float* __restrict__ Gso,
                                               float* __restrict__ Gio) {
  __shared__ float A[D * LDA]; __shared__ float V[D * LDA];
  __shared__ float f[D]; __shared__ JScratch js;
  const int tid = threadIdx.x;
  load64(A, G);
  __syncthreads();
  jacobi64(A, V, &js);
  if (tid < 64) f[tid] = sqrtf(fmaxf(A[tid * LDA + tid], 1e-20f));
  __syncthreads();
  mm64<false, true, true>(V, V, f, nullptr, Gso);
  __syncthreads();
  if (tid < 64) f[tid] = rsqrtf(fmaxf(A[tid * LDA + tid], 1e-20f));
  __syncthreads();
  mm64<false, true, true>(V, V, f, nullptr, Gio);
}

// S_b = sym(Gi X_b Gi)
__global__ __launch_bounds__(256) void k_sandwich(const float* __restrict__ Xin,
                                                  const float* __restrict__ Gi,
                                                  float* __restrict__ Sout) {
  __shared__ float Gis[D * LDA]; __shared__ float Xs[D * LDA]; __shared__ float Ts[D * LDA];
  const int b = blockIdx.x;
  load64(Gis, Gi);
  load64(Xs, Xin + (size_t)b * D * D);
  __syncthreads();
  mm64<false, false, false>(Gis, Xs, nullptr, Ts, nullptr);   // T = Gi * X
  __syncthreads();
  mm64<false, false, false>(Ts, Gis, nullptr, Xs, nullptr);   // S' = T * Gi
  __syncthreads();
  store_sym64(Sout + (size_t)b * D * D, Xs);
}

// G <- sym(Gs * expm(mean) * Gs)
__global__ __launch_bounds__(256) void k_G_update(const float* __restrict__ Mean,
                                                  const float* __restrict__ Gs,
                                                  float* __restrict__ G) {
  __shared__ float A[D * LDA]; __shared__ float V[D * LDA]; __shared__ float Es[D * LDA];
  __shared__ float f[D]; __shared__ JScratch js;
  const int tid = threadIdx.x;
  load64(A, Mean);
  __syncthreads();
  jacobi64(A, V, &js);
  if (tid < 64) f[tid] = expf(A[tid * LDA + tid]);
  __syncthreads();
  mm64<false, true, true>(V, V, f, Es, nullptr);              // E = V e^w V^T
  __syncthreads();
  load64(A, Gs);                                              // reuse A for Gs
  __syncthreads();
  mm64<false, false, false>(A, Es, nullptr, V, nullptr);      // T = Gs * E  (into V)
  __syncthreads();
  mm64<false, false, false>(V, A, nullptr, Es, nullptr);      // G' = T * Gs (into Es)
  __syncthreads();
  store_sym64(G, Es);
}

// Final tangent eig: save V, w; partial dispersion sum(log(w)^2).
__global__ __launch_bounds__(256) void k_eig4(const float* __restrict__ Sin,
                                              float* __restrict__ Vout,
                                              float* __restrict__ wout,
                                              float* __restrict__ part) {
  __shared__ float A[D * LDA]; __shared__ float V[D * LDA];
  __shared__ float red[256]; __shared__ JScratch js;
  const int tid = threadIdx.x; const int b = blockIdx.x;
  load64(A, Sin + (size_t)b * D * D);
  __syncthreads();
  jacobi64(A, V, &js);
  float lv = 0.f;
  if (tid < 64) {
    float w = fmaxf(A[tid * LDA + tid], 1e-20f);
    wout[(size_t)b * D + tid] = w;
    float l = logf(w);
    lv = l * l;
  }
  red[tid] = lv; __syncthreads();
  for (int k = 128; k > 0; k >>= 1) { if (tid < k) red[tid] += red[tid + k]; __syncthreads(); }
  if (tid == 0) part[b] = red[0];
  for (int e = tid; e < D * D; e += 256)
    Vout[(size_t)b * D * D + e] = V[(e >> 6) * LDA + (e & 63)];
}

__global__ __launch_bounds__(256) void k_reduce_v(const float* __restrict__ part,
                                                  float* __restrict__ gV) {
  __shared__ float red[256];
  const int tid = threadIdx.x;
  float s = 0.f;
  for (int b = tid; b < BATCH; b += 256) s += part[b];
  red[tid] = s; __syncthreads();
  for (int k = 128; k > 0; k >>= 1) { if (tid < k) red[tid] += red[tid + k]; __syncthreads(); }
  if (tid == 0) gV[0] = red[0] * (1.f / (float)BATCH);
}

// Y_b = Bs * (V w^s V^T) * Bs   with s = gamma / sqrt(v + eps)
__global__ __launch_bounds__(256) void k_Y(const float* __restrict__ Vin,
                                           const float* __restrict__ win,
                                           const float* __restrict__ gV,
                                           const float* __restrict__ gammap,
                                           const float* __restrict__ Bsg,
                                           float* __restrict__ Yout) {
  __shared__ float Vs[D * LDA]; __shared__ float Es[D * LDA]; __shared__ float Bss[D * LDA];
  __shared__ float f[D];
  const int tid = threadIdx.x; const int b = blockIdx.x;
  load64(Vs, Vin + (size_t)b * D * D);
  load64(Bss, Bsg);
  if (tid < 64) {
    float s = gammap[0] * rsqrtf(gV[0] + BN_EPS);
    f[tid] = expf(s * logf(fmaxf(win[(size_t)b * D + tid], 1e-20f)));
  }
  __syncthreads();
  mm64<false, true, true>(Vs, Vs, f, Es, nullptr);            // E = V w^s V^T
  __syncthreads();
  mm64<false, false, false>(Bss, Es, nullptr, Vs, nullptr);   // T = Bs * E (into Vs)
  __syncthreads();
  mm64<false, false, false>(Vs, Bss, nullptr, Es, nullptr);   // Y = T * Bs (into Es)
  __syncthreads();
  store_sym64(Yout + (size_t)b * D * D, Es);
}

// LogEig + upper-triangle vectorization + Wc projection -> out (B, 4)
__global__ __launch_bounds__(256) void k_eig5(const float* __restrict__ Yin,
                                              const float* __restrict__ Wc,
                                              const float* __restrict__ bc,
                                              float* __restrict__ out) {
  __shared__ float A[D * LDA]; __shared__ float V[D * LDA]; __shared__ float Rs[D * LDA];
  __shared__ float f[D]; __shared__ float red[256]; __shared__ JScratch js;
  const int tid = threadIdx.x; const int b = blockIdx.x;
  load64(A, Yin + (size_t)b * D * D);
  __syncthreads();
  jacobi64(A, V, &js);
  if (tid < 64) f[tid] = logf(fmaxf(A[tid * LDA + tid], 1e-20f));
  __syncthreads();
  mm64<false, true, true>(V, V, f, Rs, nullptr);              // Lg = V log(w) V^T
  __syncthreads();
  float p[4] = {0.f, 0.f, 0.f, 0.f};
  for (int e = tid; e < D * D; e += 256) {
    int i = e >> 6, j = e & 63;
    if (j >= i) {
      int l = i * D - (i * (i - 1)) / 2 + (j - i);            // row-major triu index
      float cf = (i == j) ? 1.f : SQRT2F;
      float val = Rs[i * LDA + j] * cf;
#pragma unroll
      for (int o = 0; o < 4; ++o) p[o] += val * Wc[o * 2080 + l];
    }
  }
#pragma unroll
  for (int o = 0; o < 4; ++o) {
    red[tid] = p[o]; __syncthreads();
    for (int k = 128; k > 0; k >>= 1) { if (tid < k) red[tid] += red[tid + k]; __syncthreads(); }
    if (tid == 0) out[(size_t)b * 4 + o] = red[0] + bc[o];
    __syncthreads();
  }
}

// ---------------------------------------------------------------------------
extern "C" void kernel_launch(void* const* d_in, const int* in_sizes, int n_in,
                              void* d_out, int out_size, void* d_ws, size_t ws_size,
                              hipStream_t stream) {
  (void)in_sizes; (void)n_in; (void)out_size; (void)ws_size;
  const float* X     = (const float*)d_in[0];
  const float* alpha = (const float*)d_in[1];
  const float* W     = (const float*)d_in[2];
  const float* gamma = (const float*)d_in[3];
  const float* bnb   = (const float*)d_in[4];
  const float* Wc    = (const float*)d_in[5];
  const float* bc    = (const float*)d_in[6];
  float* out = (float*)d_out;

  const size_t MAT = (size_t)D * D;
  float* buf0  = (float*)d_ws;               // M / S / Y      (BATCH x 64 x 64)
  float* buf1  = buf0 + (size_t)BATCH * MAT; // X after ReEig
  float* buf2  = buf1 + (size_t)BATCH * MAT; // log-mats / V
  float* wbuf  = buf2 + (size_t)BATCH * MAT; // eigenvalues    (BATCH x 64)
  float* part  = wbuf + (size_t)BATCH * D;   // per-block dispersion partials
  float* gWtW  = part + BATCH;
  float* gBs   = gWtW + MAT;
  float* gG    = gBs + MAT;
  float* gGs   = gG + MAT;
  float* gGi   = gGs + MAT;
  float* gMean = gGi + MAT;
  float* gV    = gMean + MAT;

  k_prep<<<1, 256, 0, stream>>>(W, bnb, gWtW, gBs);
  k_bimap<<<BATCH, 256, 0, stream>>>(X, alpha, W, gWtW, buf0);
  k_eig_reeig<<<BATCH, 256, 0, stream>>>(buf0, buf1, buf2);
  k_reduce_mean<<<D * D, 256, 0, stream>>>(buf2, gMean);
  k_expm_small<<<1, 256, 0, stream>>>(gMean, gG);
  for (int it = 0; it < 2; ++it) {
    k_G_pow<<<1, 256, 0, stream>>>(gG, gGs, gGi);
    k_sandwich<<<BATCH, 256, 0, stream>>>(buf1, gGi, buf0);
    k_eig_log<<<BATCH, 256, 0, stream>>>(buf0, buf2);
    k_reduce_mean<<<D * D, 256, 0, stream>>>(buf2, gMean);
    k_G_update<<<1, 256, 0, stream>>>(gMean, gGs, gG);
  }
  k_G_pow<<<1, 256, 0, stream>>>(gG, gGs, gGi);
  k_sandwich<<<BATCH, 256, 0, stream>>>(buf1, gGi, buf0);
  k_eig4<<<BATCH, 256, 0, stream>>>(buf0, buf2, wbuf, part);
  k_reduce_v<<<1, 256, 0, stream>>>(part, gV);
  k_Y<<<BATCH, 256, 0, stream>>>(buf2, wbuf, gV, gamma, gBs, buf0);
  k_eig5<<<BATCH, 256, 0, stream>>>(buf0, Wc, bc, out);
}